// Modularity_79860621902560
// MI455X (gfx1250) — compile-verified
//
#include <hip/hip_runtime.h>

// ---------------------------------------------------------------------------
// Modularity GCN + soft-kmeans pipeline for MI455X (gfx1250, wave32).
// Dense GEMMs use V_WMMA_F32_16X16X4_F32 (fp32 matrix pipe, exact precision).
// B operands are consumed transposed ([Nc,Kd] row-major) so every fragment
// load is a single global_load_b64; weights are pre-transposed once.
// Each wave owns a 16x(16*NT) output tile: one A fragment feeds NT WMMAs
// per k-step. Loads are batched ahead of the WMMA group so they clause and
// the loadcnt waits stage instead of serializing each WMMA.
// ---------------------------------------------------------------------------

typedef float v2f __attribute__((ext_vector_type(2)));
typedef float v8f __attribute__((ext_vector_type(8)));

#define CLUSTER_TEMP 5.0f
#define EPS_V 1e-8f

// ---------------------------------------------------------------------------
__global__ void mod_zero_f32(float* __restrict__ p, long n) {
  long i = (long)blockIdx.x * blockDim.x + threadIdx.x;
  if (i < n) p[i] = 0.0f;
}

// ---------------------------------------------------------------------------
// Wt[Nc,Kd] = W[Kd,Nc]^T  (tiny weight matrices; one thread per element)
// ---------------------------------------------------------------------------
__global__ void mod_transpose(const float* __restrict__ W, float* __restrict__ Wt,
                              int Kd, int Nc) {
  int i = blockIdx.x * blockDim.x + threadIdx.x;
  if (i >= Kd * Nc) return;
  int k = i / Nc;
  int c = i - k * Nc;
  Wt[(size_t)c * Kd + k] = W[i];
}

// ---------------------------------------------------------------------------
// C[M,Nc] = A[M,Kd] @ Bt^T, with Bt row-major [Nc,Kd]. One wave per
// 16x(16*NT) output tile, K-step 4 (V_WMMA_F32_16X16X4_F32).
//
// f32 A fragment (16x4, 2 VGPRs): lanes 0-15: M=lane,    K={k,k+1};
//                                 lanes 16-31: M=lane-16, K={k+2,k+3}.
// f32 B fragment (4x16, 2 VGPRs): lanes 0-15: N=lane,    K={k,k+1};
//                                 lanes 16-31: N=lane-16, K={k+2,k+3}.
// f32 C/D (16x16, 8 VGPRs): VGPR v -> row M = v + 8*(lane>=16), N = lane%16.
// Requires M%16==0, Kd%4==0, Nc%(16*NT)==0.
// ---------------------------------------------------------------------------
template <int NT>
__global__ void mod_gemm_wmma_f32(const float* __restrict__ A,
                                  const float* __restrict__ Bt,
                                  float* __restrict__ C,
                                  int M, int Kd, int Nc) {
  int wavesPerBlock = blockDim.x >> 5;
  int wave = blockIdx.x * wavesPerBlock + (threadIdx.x >> 5);
  int ntN = Nc / (16 * NT);                // tiles along N
  int ntiles = (M >> 4) * ntN;
  if (wave >= ntiles) return;              // wave-uniform: EXEC stays all-1s

  int mt = wave / ntN;
  int nt = wave - mt * ntN;
  int lane = threadIdx.x & 31;
  int half = lane >> 4;                    // 0: K pair {k,k+1}; 1: {k+2,k+3}
  int l    = lane & 15;

  const float* Arow = A + (size_t)(mt * 16 + l) * Kd + half * 2;
  const float* Bp[NT];
  #pragma unroll
  for (int t = 0; t < NT; ++t)
    Bp[t] = Bt + (size_t)(nt * 16 * NT + t * 16 + l) * Kd + half * 2;

  v8f acc[NT];
  #pragma unroll
  for (int t = 0; t < NT; ++t) acc[t] = (v8f){};

  #pragma unroll 4
  for (int k = 0; k < Kd; k += 4) {
    // Batch all fragment loads first so they issue as a clause and the
    // loadcnt waits stage across the WMMA group (MLP inside the wave).
    v2f a = *(const v2f*)(Arow + k);       // global_load_b64 (8B aligned)
    v2f b[NT];
    #pragma unroll
    for (int t = 0; t < NT; ++t)
      b[t] = *(const v2f*)(Bp[t] + k);     // global_load_b64 (8B aligned)
    #pragma unroll
    for (int t = 0; t < NT; ++t)
      acc[t] = __builtin_amdgcn_wmma_f32_16x16x4_f32(
          false, a, false, b[t], (short)0, acc[t], false, false);
  }

  float* Crow = C + (size_t)(mt * 16 + half * 8) * Nc + nt * 16 * NT + l;
  #pragma unroll
  for (int v = 0; v < 8; ++v) {
    #pragma unroll
    for (int t = 0; t < NT; ++t)
      Crow[(size_t)v * Nc + t * 16] = acc[t][v];
  }
}

// ---------------------------------------------------------------------------
// acc[dst[e]*F + f] += w[e] * sup[src[e]*F + f]   (thread per (edge, feature))
// ---------------------------------------------------------------------------
__global__ void mod_edge_scatter(const int* __restrict__ src,
                                 const int* __restrict__ dst,
                                 const float* __restrict__ w,
                                 const float* __restrict__ sup,
                                 float* __restrict__ acc,
                                 int F, long total) {
  long idx = (long)blockIdx.x * blockDim.x + threadIdx.x;
  if (idx >= total) return;
  int e = (int)(idx / F);
  int f = (int)(idx - (long)e * F);
  float v = w[e] * sup[(size_t)src[e] * F + f];
  atomicAdd(&acc[(size_t)dst[e] * F + f], v);
}

// ---------------------------------------------------------------------------
__global__ void mod_bias_relu(float* __restrict__ h, const float* __restrict__ b,
                              long n, int F) {
  long i = (long)blockIdx.x * blockDim.x + threadIdx.x;
  if (i >= n) return;
  float v = h[i] + b[i % F];
  h[i] = v > 0.0f ? v : 0.0f;
}

// ---------------------------------------------------------------------------
// embeds = acc + b2 (written to output); data = row-normalized embeds.
// One wave (32 lanes) per 64-wide row; lane handles f and f+32.
// ---------------------------------------------------------------------------
__global__ void mod_embeds_norm(const float* __restrict__ acc,
                                const float* __restrict__ b2,
                                float* __restrict__ embeds,
                                float* __restrict__ data, int Nn) {
  int wavesPerBlock = blockDim.x >> 5;
  int row = blockIdx.x * wavesPerBlock + (threadIdx.x >> 5);
  if (row >= Nn) return;
  int lane = threadIdx.x & 31;
  size_t base = (size_t)row * 64;
  float e0 = acc[base + lane]      + b2[lane];
  float e1 = acc[base + lane + 32] + b2[lane + 32];
  embeds[base + lane]      = e0;
  embeds[base + lane + 32] = e1;
  float ss = e0 * e0 + e1 * e1;
  #pragma unroll
  for (int m = 16; m >= 1; m >>= 1) ss += __shfl_xor(ss, m, 32);
  float inv = rsqrtf(ss);
  data[base + lane]      = e0 * inv;
  data[base + lane + 32] = e1 * inv;
}

// ---------------------------------------------------------------------------
// Per node (one wave, lane = cluster k): r = softmax(TEMP * dist_row).
// Optionally writes r; optionally accumulates cluster_r (r_sum[K]) and
// r^T @ data (mu_num[K,64]) via LDS block partials + global atomics.
// ---------------------------------------------------------------------------
__global__ void mod_softmax_accum(const float* __restrict__ dist,
                                  const float* __restrict__ data,
                                  float* __restrict__ r_out,
                                  float* __restrict__ r_sum,
                                  float* __restrict__ mu_num,
                                  int Nn, int writeR, int accum) {
  __shared__ float s_rsum[32];
  __shared__ float s_mnum[32 * 64];
  if (accum) {
    for (int i = threadIdx.x; i < 32 * 64; i += blockDim.x) s_mnum[i] = 0.0f;
    if (threadIdx.x < 32) s_rsum[threadIdx.x] = 0.0f;
    __syncthreads();
  }

  int wavesPerBlock = blockDim.x >> 5;
  int row = blockIdx.x * wavesPerBlock + (threadIdx.x >> 5);
  int lane = threadIdx.x & 31;

  if (row < Nn) {
    float val = CLUSTER_TEMP * dist[(size_t)row * 32 + lane];
    float m = val;
    #pragma unroll
    for (int i = 16; i >= 1; i >>= 1) m = fmaxf(m, __shfl_xor(m, i, 32));
    float ex = __expf(val - m);
    float s = ex;
    #pragma unroll
    for (int i = 16; i >= 1; i >>= 1) s += __shfl_xor(s, i, 32);
    float r = ex / s;
    if (writeR) r_out[(size_t)row * 32 + lane] = r;
    if (accum) {
      atomicAdd(&s_rsum[lane], r);
      const float* drow = data + (size_t)row * 64;
      for (int d = 0; d < 64; ++d)
        atomicAdd(&s_mnum[lane * 64 + d], r * drow[d]);
    }
  }

  if (accum) {
    __syncthreads();
    if (threadIdx.x < 32) atomicAdd(&r_sum[threadIdx.x], s_rsum[threadIdx.x]);
    for (int i = threadIdx.x; i < 32 * 64; i += blockDim.x)
      atomicAdd(&mu_num[i], s_mnum[i]);
  }
}

// ---------------------------------------------------------------------------
__global__ void mod_mu_update(float* __restrict__ mu,
                              const float* __restrict__ mu_num,
                              const float* __restrict__ r_sum, int n) {
  int i = blockIdx.x * blockDim.x + threadIdx.x;
  if (i < n) mu[i] = mu_num[i] / (r_sum[i >> 6] + EPS_V);
}

// ---------------------------------------------------------------------------
static inline unsigned cdiv_l(long a, long b) { return (unsigned)((a + b - 1) / b); }

extern "C" void kernel_launch(void* const* d_in, const int* in_sizes, int n_in,
                              void* d_out, int out_size, void* d_ws, size_t ws_size,
                              hipStream_t stream) {
  const float* x   = (const float*)d_in[0];
  const int*   src = (const int*)d_in[1];
  const int*   dst = (const int*)d_in[2];
  const float* ew  = (const float*)d_in[3];
  const float* W1  = (const float*)d_in[4];
  const float* b1  = (const float*)d_in[5];
  const float* W2  = (const float*)d_in[6];
  const float* b2  = (const float*)d_in[7];
  const float* mu0 = (const float*)d_in[8];

  const int NFEAT = 256, NHID = 128, NOUT = 64, K = 32;
  const int N = in_sizes[0] / NFEAT;   // 50000 (divisible by 16)
  const int E = in_sizes[1];           // 800000
  const int NUM_UPDATES = 6;           // 2 cluster_fn calls x num_iter(3)

  // ---- workspace layout (floats), with lifetime-based aliasing -------------
  float* ws = (float*)d_ws;
  size_t off = 0;
  float* bufA = ws + off; off += (size_t)N * NHID;  // support1 | support2 + eacc
  float* bufB = ws + off; off += (size_t)N * NHID;  // h        | data + dist
  float* mu_cur = ws + off; off += (size_t)K * NOUT;
  float* mu_num = ws + off; off += (size_t)K * NOUT;  // mu_num, r_sum adjacent
  float* r_sum  = ws + off; off += (size_t)K;
  float* W1t    = ws + off; off += (size_t)NFEAT * NHID;  // W1^T [NHID,NFEAT]
  float* W2t    = ws + off; off += (size_t)NHID * NOUT;   // W2^T [NOUT,NHID]

  float* support1 = bufA;
  float* h        = bufB;
  float* support2 = bufA;                         // support1 dead by then
  float* eacc     = bufA + (size_t)N * NOUT;
  float* data     = bufB;                         // h dead by then
  float* dist_ws  = bufB + (size_t)N * NOUT;

  // ---- output layout: mu_out[K,64], r[N,32], embeds[N,64], dist[N,32] ------
  float* out      = (float*)d_out;
  float* out_mu   = out;
  float* out_r    = out_mu + (size_t)K * NOUT;
  float* out_emb  = out_r  + (size_t)N * K;
  float* out_dist = out_emb + (size_t)N * NOUT;

  dim3 blk(256);

  // 0) transpose weights (tiny)
  mod_transpose<<<cdiv_l(NFEAT * NHID, 256), blk, 0, stream>>>(W1, W1t, NFEAT, NHID);
  mod_transpose<<<cdiv_l(NHID * NOUT, 256), blk, 0, stream>>>(W2, W2t, NHID, NOUT);

  // 1) support1 = x @ W1                      [WMMA f32, 16x64 tiles]
  mod_gemm_wmma_f32<4><<<cdiv_l((long)(N / 16) * (NHID / 64), 8), blk, 0, stream>>>(
      x, W1t, support1, N, NFEAT, NHID);

  // 2) h = relu(spmm(support1) + b1)
  mod_zero_f32<<<cdiv_l((long)N * NHID, 256), blk, 0, stream>>>(h, (long)N * NHID);
  mod_edge_scatter<<<cdiv_l((long)E * NHID, 256), blk, 0, stream>>>(
      src, dst, ew, support1, h, NHID, (long)E * NHID);
  mod_bias_relu<<<cdiv_l((long)N * NHID, 256), blk, 0, stream>>>(
      h, b1, (long)N * NHID, NHID);

  // 3) support2 = h @ W2                      [WMMA f32, 16x64 tiles]
  mod_gemm_wmma_f32<4><<<cdiv_l((long)(N / 16) * (NOUT / 64), 8), blk, 0, stream>>>(
      h, W2t, support2, N, NHID, NOUT);

  // 4) eacc = spmm(support2)
  mod_zero_f32<<<cdiv_l((long)N * NOUT, 256), blk, 0, stream>>>(eacc, (long)N * NOUT);
  mod_edge_scatter<<<cdiv_l((long)E * NOUT, 256), blk, 0, stream>>>(
      src, dst, ew, support2, eacc, NOUT, (long)E * NOUT);

  // 5) embeds = eacc + b2 -> output; data = row-normalized embeds
  mod_embeds_norm<<<cdiv_l(N, 8), blk, 0, stream>>>(eacc, b2, out_emb, data, N);

  // 6) soft-kmeans: 6 mu updates starting from mu0
  //    mu [K,NOUT] is already the transposed-B layout for dist = data @ mu^T.
  hipMemcpyAsync(mu_cur, mu0, (size_t)K * NOUT * sizeof(float),
                 hipMemcpyDeviceToDevice, stream);
  for (int it = 0; it < NUM_UPDATES; ++it) {
    mod_zero_f32<<<cdiv_l((long)K * NOUT + K, 256), blk, 0, stream>>>(
        mu_num, (long)K * NOUT + K);                 // mu_num + r_sum
    mod_gemm_wmma_f32<2><<<cdiv_l((long)(N / 16) * (K / 32), 8), blk, 0, stream>>>(
        data, mu_cur, dist_ws, N, NOUT, K);          // dist [WMMA, 16x32 tiles]
    mod_softmax_accum<<<cdiv_l(N, 8), blk, 0, stream>>>(
        dist_ws, data, nullptr, r_sum, mu_num, N, 0, 1);
    mod_mu_update<<<cdiv_l((long)K * NOUT, 256), blk, 0, stream>>>(
        mu_cur, mu_num, r_sum, K * NOUT);
  }

  // 7) outputs: mu_out, final dist, final r
  hipMemcpyAsync(out_mu, mu_cur, (size_t)K * NOUT * sizeof(float),
                 hipMemcpyDeviceToDevice, stream);
  mod_gemm_wmma_f32<2><<<cdiv_l((long)(N / 16) * (K / 32), 8), blk, 0, stream>>>(
      data, mu_cur, out_dist, N, NOUT, K);           // [WMMA]
  mod_softmax_accum<<<cdiv_l(N, 8), blk, 0, stream>>>(
      out_dist, data, out_r, nullptr, nullptr, N, 1, 0);
}